// GINNet_geom_11269994184788
// MI455X (gfx1250) — compile-verified
//
#include <hip/hip_runtime.h>
#include <hip/hip_bf16.h>

#define D 128
#define LDS_STRIDE 130  // pad to avoid 64-bank conflicts (stride%64==2, even for v2f align)

typedef __attribute__((ext_vector_type(2))) float v2f;
typedef __attribute__((ext_vector_type(8))) float v8f;

// ---------------------------------------------------------------------------
// agg[i] = (1 + eps) * x[i]   (elementwise init of the aggregation buffer)
// ---------------------------------------------------------------------------
__global__ __launch_bounds__(256) void gin_init_kernel(
    const float* __restrict__ x, const float* __restrict__ epsPtr,
    float* __restrict__ agg, int nElem) {
  int i = blockIdx.x * blockDim.x + threadIdx.x;
  if (i >= nElem) return;
  float scale = 1.0f + *epsPtr;
  agg[i] = scale * x[i];
}

// ---------------------------------------------------------------------------
// agg[dst] += x[src] for every edge. One 32-lane wave handles one edge:
// each lane moves a float4 chunk (32*16B = 512B = one full feature row).
// Whole working set (25.6MB x2) lives in the 192MB L2, so atomics resolve
// at L2 without touching HBM after first fill.
// ---------------------------------------------------------------------------
__global__ __launch_bounds__(256) void gin_scatter_kernel(
    const float* __restrict__ x, const long long* __restrict__ ei,
    float* __restrict__ agg, int nEdges) {
  int tid = blockIdx.x * blockDim.x + threadIdx.x;
  int e = tid >> 5;          // edge id (32 lanes per edge)
  int c = (tid & 31) << 2;   // float4 chunk offset within the 128-dim row
  if (e >= nEdges) return;
  long long s = ei[e];           // edge_index[0, e]
  long long d = ei[nEdges + e];  // edge_index[1, e]
  const float4 v = *(const float4*)(x + s * (long long)D + c);
  float* p = agg + d * (long long)D + c;
  __hip_atomic_fetch_add(p + 0, v.x, __ATOMIC_RELAXED, __HIP_MEMORY_SCOPE_AGENT);
  __hip_atomic_fetch_add(p + 1, v.y, __ATOMIC_RELAXED, __HIP_MEMORY_SCOPE_AGENT);
  __hip_atomic_fetch_add(p + 2, v.z, __ATOMIC_RELAXED, __HIP_MEMORY_SCOPE_AGENT);
  __hip_atomic_fetch_add(p + 3, v.w, __ATOMIC_RELAXED, __HIP_MEMORY_SCOPE_AGENT);
}

// ---------------------------------------------------------------------------
// Y = act(X @ W^T + b)   with X:[N,128], W:[128,128] row-major, b:[128]
// Full-precision fp32 WMMA path: V_WMMA_F32_16X16X4_F32.
// Block = 256 threads = 8 wave32; W staged in LDS once; each wave computes a
// 16x128 output slab (8 accumulator tiles, 32 k-steps, 256 WMMAs).
// ---------------------------------------------------------------------------
__global__ __launch_bounds__(256) void gin_gemm_kernel(
    const float* __restrict__ X, const float* __restrict__ W,
    const float* __restrict__ bias, float* __restrict__ Y,
    int N, int doRelu) {
  __shared__ float ldsW[D * LDS_STRIDE];  // ~66.6KB (WGP has 320KB LDS)

  const int t = threadIdx.x;
  // Cooperative, coalesced copy of W into padded LDS.
  for (int r = t / 32; r < D; r += 8) {        // 8 waves, 8 rows each pass (16 per wave)
    int lane = t & 31;
    for (int k4 = lane; k4 < D / 4; k4 += 32) {
      float4 w = *(const float4*)(W + r * D + k4 * 4);
      float* dst = &ldsW[r * LDS_STRIDE + k4 * 4];
      dst[0] = w.x; dst[1] = w.y; dst[2] = w.z; dst[3] = w.w;
    }
  }
  __syncthreads();

  const int wave = t >> 5;
  const int lane = t & 31;
  const int row0 = (blockIdx.x * 8 + wave) * 16;  // 16 rows per wave
  if (row0 >= N) return;

  const int m = lane & 15;            // M (lanes 0-15) / M (lanes 16-31)
  const int khalf = (lane >> 4) * 2;  // lanes 16-31 carry K+2, K+3

  v8f acc[8];
#pragma unroll
  for (int nt = 0; nt < 8; ++nt) acc[nt] = (v8f){};

  int arow = row0 + m;
  if (arow >= N) arow = N - 1;  // clamp (stores are guarded below)
  const float* aptr = X + (long long)arow * D;

#pragma unroll 4
  for (int kk = 0; kk < 32; ++kk) {
    const int kb = kk * 4 + khalf;
    // A fragment: A[M=m, K=kb..kb+1] per ISA 32-bit 16x4 layout
    v2f a = *(const v2f*)(aptr + kb);
#pragma unroll
    for (int nt = 0; nt < 8; ++nt) {
      // B = W^T: B[K=kb..kb+1, N=nt*16+m] = W[n, kb..kb+1]
      const int n = nt * 16 + m;
      v2f b = *(const v2f*)(&ldsW[n * LDS_STRIDE + kb]);
      acc[nt] = __builtin_amdgcn_wmma_f32_16x16x4_f32(
          false, a, false, b, (short)0, acc[nt], false, false);
    }
  }

  // Epilogue: C/D layout -> VGPR r holds M=r (lanes 0-15) or M=r+8 (16-31)
  const int mOff = (lane >> 4) * 8;
#pragma unroll
  for (int nt = 0; nt < 8; ++nt) {
    const int col = nt * 16 + m;
    const float bv = bias[col];
#pragma unroll
    for (int r = 0; r < 8; ++r) {
      const int row = row0 + r + mOff;
      if (row < N) {
        float v = acc[nt][r] + bv;
        if (doRelu) v = fmaxf(v, 0.0f);
        Y[(long long)row * D + col] = v;
      }
    }
  }
}

// ---------------------------------------------------------------------------
// Pipeline:
//   agg1 = (1+eps1)*x + scatter(x)          -> buf0
//   t1   = relu(agg1 @ w1a^T + b1a)         -> buf1
//   x1   = relu(t1 @ w1b^T + b1b)           -> buf0   (outer relu folded in)
//   agg2 = (1+eps2)*x1 + scatter(x1)        -> buf1
//   t2   = relu(agg2 @ w2a^T + b2a)         -> buf0
//   out  = t2 @ w2b^T + b2b                 -> d_out
// ---------------------------------------------------------------------------
extern "C" void kernel_launch(void* const* d_in, const int* in_sizes, int n_in,
                              void* d_out, int out_size, void* d_ws, size_t ws_size,
                              hipStream_t stream) {
  const float*     feats = (const float*)d_in[0];
  const long long* ei    = (const long long*)d_in[1];
  const float* w1a = (const float*)d_in[2];
  const float* b1a = (const float*)d_in[3];
  const float* w1b = (const float*)d_in[4];
  const float* b1b = (const float*)d_in[5];
  const float* eps1 = (const float*)d_in[6];
  const float* w2a = (const float*)d_in[7];
  const float* b2a = (const float*)d_in[8];
  const float* w2b = (const float*)d_in[9];
  const float* b2b = (const float*)d_in[10];
  const float* eps2 = (const float*)d_in[11];

  const int N = in_sizes[0] / D;       // 50000
  const int E = in_sizes[1] / 2;       // 800000
  const int nElem = N * D;

  float* buf0 = (float*)d_ws;
  float* buf1 = buf0 + (size_t)nElem;
  float* out  = (float*)d_out;

  const int initBlocks    = (nElem + 255) / 256;
  const int scatterBlocks = (int)(((long long)E * 32 + 255) / 256);
  const int gemmBlocks    = ((N + 15) / 16 + 7) / 8;

  // ---- layer 1 ----
  gin_init_kernel<<<initBlocks, 256, 0, stream>>>(feats, eps1, buf0, nElem);
  gin_scatter_kernel<<<scatterBlocks, 256, 0, stream>>>(feats, ei, buf0, E);
  gin_gemm_kernel<<<gemmBlocks, 256, 0, stream>>>(buf0, w1a, b1a, buf1, N, 1);
  gin_gemm_kernel<<<gemmBlocks, 256, 0, stream>>>(buf1, w1b, b1b, buf0, N, 1);

  // ---- layer 2 ----
  gin_init_kernel<<<initBlocks, 256, 0, stream>>>(buf0, eps2, buf1, nElem);
  gin_scatter_kernel<<<scatterBlocks, 256, 0, stream>>>(buf0, ei, buf1, E);
  gin_gemm_kernel<<<gemmBlocks, 256, 0, stream>>>(buf1, w2a, b2a, buf0, N, 1);
  gin_gemm_kernel<<<gemmBlocks, 256, 0, stream>>>(buf0, w2b, b2b, out, N, 0);
}